// SPMotifNet_72009421684760
// MI455X (gfx1250) — compile-verified
//
#include <hip/hip_runtime.h>

// ---------------------------------------------------------------------------
// SPMotif LEConv GNN for gfx1250 (MI455X).
// GEMMs use V_WMMA_F32_16X16X4_F32 (native fp32 WMMA -> exact fp32 semantics).
// Weights are pre-transposed once so B fragments load as single b64; the
// K-loop is manually double-buffered so fragment loads for step k+4 issue as
// a clause while step k's WMMAs execute (avoids load->wait0->wmma serialization
// seen when the compiler recycles one fragment register pair).
// Edge scatter uses relaxed agent-scope float atomics (working set ~103MB is
// L2-resident: 192MB L2).
// ---------------------------------------------------------------------------

typedef __attribute__((ext_vector_type(2))) float v2f;
typedef __attribute__((ext_vector_type(8))) float v8f;

#define NNODES  100000
#define NEDGES  1600000
#define NGRAPHS 512
#define XDIM    4
#define HDIM    64
#define LAYERS  3
#define NCLASS  3

__global__ void zero_kernel(float* __restrict__ p, int n) {
    int i = blockIdx.x * blockDim.x + threadIdx.x;
    if (i < n) p[i] = 0.0f;
}

// Transpose all 9 [64,64] layer weights into wt[(l*3+which)*4096 + n*64 + k].
__global__ void transpose_w_kernel(const float* __restrict__ W1,
                                   const float* __restrict__ W2,
                                   const float* __restrict__ W3,
                                   float* __restrict__ wt) {
    int i = blockIdx.x * blockDim.x + threadIdx.x;
    if (i >= LAYERS * 3 * HDIM * HDIM) return;
    int m   = i >> 12;     // matrix 0..8  (= l*3 + which)
    int rem = i & 4095;
    int n   = rem >> 6;    // output-major index = column of W
    int k   = rem & 63;
    int l = m / 3, which = m % 3;
    const float* src = (which == 0 ? W1 : which == 1 ? W2 : W3) + l * HDIM * HDIM;
    wt[m * HDIM * HDIM + n * HDIM + k] = src[k * HDIM + n];
}

// h = x @ W_emb + b_emb   (K=4, VALU is fine)
__global__ void embed_kernel(const float* __restrict__ x,
                             const float* __restrict__ W,
                             const float* __restrict__ bias,
                             float* __restrict__ h) {
    int i = blockIdx.x * blockDim.x + threadIdx.x;
    if (i >= NNODES * HDIM) return;
    int n = i >> 6, j = i & 63;
    const float* xr = x + n * XDIM;
    float acc = bias[j];
#pragma unroll
    for (int k = 0; k < XDIM; ++k) acc += xr[k] * W[k * HDIM + j];
    h[i] = acc;
}

// ---------------------------------------------------------------------------
// V_WMMA_F32_16X16X4_F32 fragments (wave32):
// A (16x4): lane&15 = M row; VGPR0 = K = k+2*(lane>>4), VGPR1 = K+1 -> b64.
// B (4x16): lane&15 = N col; with W transposed the (K,K+1) pair is contiguous
//           -> single b64.
// C/D (16x16 f32): lanes 0-15 N=lane, VGPR r -> M=r; lanes 16-31 M=8+r.
// ---------------------------------------------------------------------------

// a = h@W1 + b1 ; b = h@W2   -- fused: h rows read once per tile.
__global__ void __launch_bounds__(256)
gemm2_kernel(const float* __restrict__ A,
             const float* __restrict__ Wt1, const float* __restrict__ b1,
             const float* __restrict__ Wt2,
             float* __restrict__ outA, float* __restrict__ outB) {
    const int lane = threadIdx.x & 31;
    const int wave = (blockIdx.x * blockDim.x + threadIdx.x) >> 5;
    const int row0 = wave * 16;
    if (row0 >= NNODES) return;
    const int half = lane >> 4;
    const int rc   = lane & 15;     // A: row within tile; B & C/D: column

    v8f accA[4] = {};
    v8f accB[4] = {};

    int arow = row0 + rc;
    if (arow >= NNODES) arow = NNODES - 1;
    const float* ap  = A   + arow * HDIM + 2 * half;
    const float* bp1 = Wt1 + rc * HDIM + 2 * half;   // + t*16*HDIM + k
    const float* bp2 = Wt2 + rc * HDIM + 2 * half;

    // double-buffered fragments: distinct registers per parity
    v2f af[2], f1[2][4], f2[2][4];
    af[0] = *(const v2f*)(ap);
#pragma unroll
    for (int t = 0; t < 4; ++t) {
        f1[0][t] = *(const v2f*)(bp1 + t * 16 * HDIM);
        f2[0][t] = *(const v2f*)(bp2 + t * 16 * HDIM);
    }

#pragma unroll
    for (int ks = 0; ks < 16; ++ks) {
        const int cur = ks & 1, nxt = cur ^ 1;
        if (ks < 15) {
            const int kn = (ks + 1) * 4;
            af[nxt] = *(const v2f*)(ap + kn);
#pragma unroll
            for (int t = 0; t < 4; ++t) {
                f1[nxt][t] = *(const v2f*)(bp1 + t * 16 * HDIM + kn);
                f2[nxt][t] = *(const v2f*)(bp2 + t * 16 * HDIM + kn);
            }
        }
#pragma unroll
        for (int t = 0; t < 4; ++t) {
            accA[t] = __builtin_amdgcn_wmma_f32_16x16x4_f32(
                false, af[cur], false, f1[cur][t], (short)0, accA[t], false, false);
            accB[t] = __builtin_amdgcn_wmma_f32_16x16x4_f32(
                false, af[cur], false, f2[cur][t], (short)0, accB[t], false, false);
        }
    }

    const int mb = 8 * half;
#pragma unroll
    for (int t = 0; t < 4; ++t) {
        float bias = b1[t * 16 + rc];
#pragma unroll
        for (int r = 0; r < 8; ++r) {
            int row = row0 + mb + r;
            if (row < NNODES) {
                int idx = row * HDIM + t * 16 + rc;
                outA[idx] = accA[t][r] + bias;
                outB[idx] = accB[t][r];
            }
        }
    }
}

// h = relu(agg + h@W3 + b3), in place (each wave writes only the rows it read).
__global__ void __launch_bounds__(256)
gemm3_relu_kernel(float* __restrict__ h,
                  const float* __restrict__ Wt3, const float* __restrict__ b3,
                  const float* __restrict__ agg) {
    const int lane = threadIdx.x & 31;
    const int wave = (blockIdx.x * blockDim.x + threadIdx.x) >> 5;
    const int row0 = wave * 16;
    if (row0 >= NNODES) return;
    const int half = lane >> 4;
    const int rc   = lane & 15;

    v8f acc[4] = {};

    int arow = row0 + rc;
    if (arow >= NNODES) arow = NNODES - 1;
    const float* ap = h   + arow * HDIM + 2 * half;
    const float* bp = Wt3 + rc * HDIM + 2 * half;

    v2f af[2], bf[2][4];
    af[0] = *(const v2f*)(ap);
#pragma unroll
    for (int t = 0; t < 4; ++t) bf[0][t] = *(const v2f*)(bp + t * 16 * HDIM);

#pragma unroll
    for (int ks = 0; ks < 16; ++ks) {
        const int cur = ks & 1, nxt = cur ^ 1;
        if (ks < 15) {
            const int kn = (ks + 1) * 4;
            af[nxt] = *(const v2f*)(ap + kn);
#pragma unroll
            for (int t = 0; t < 4; ++t)
                bf[nxt][t] = *(const v2f*)(bp + t * 16 * HDIM + kn);
        }
#pragma unroll
        for (int t = 0; t < 4; ++t) {
            acc[t] = __builtin_amdgcn_wmma_f32_16x16x4_f32(
                false, af[cur], false, bf[cur][t], (short)0, acc[t], false, false);
        }
    }

    const int mb = 8 * half;
#pragma unroll
    for (int t = 0; t < 4; ++t) {
        float bias = b3[t * 16 + rc];
#pragma unroll
        for (int r = 0; r < 8; ++r) {
            int row = row0 + mb + r;
            if (row < NNODES) {
                int idx = row * HDIM + t * 16 + rc;
                float v = acc[t][r] + agg[idx] + bias;
                h[idx] = fmaxf(v, 0.0f);
            }
        }
    }
}

// agg[dst] += (a[src] - b[dst]) * edge_attr ; 16 lanes/edge, float4 per lane.
__global__ void edge_kernel(const int* __restrict__ ei,
                            const float* __restrict__ ea,
                            const float* __restrict__ a,
                            const float* __restrict__ b,
                            float* __restrict__ agg) {
    int tid = blockIdx.x * blockDim.x + threadIdx.x;
    if (tid >= NEDGES * 16) return;
    int e = tid >> 4;
    int j = (tid & 15) * 4;
    int s = ei[e];
    int d = ei[NEDGES + e];
    float w = ea[e];
    const float4 av = *(const float4*)(a + s * HDIM + j);
    const float4 bv = *(const float4*)(b + d * HDIM + j);
    float* p = agg + d * HDIM + j;
    __hip_atomic_fetch_add(p + 0, (av.x - bv.x) * w, __ATOMIC_RELAXED, __HIP_MEMORY_SCOPE_AGENT);
    __hip_atomic_fetch_add(p + 1, (av.y - bv.y) * w, __ATOMIC_RELAXED, __HIP_MEMORY_SCOPE_AGENT);
    __hip_atomic_fetch_add(p + 2, (av.z - bv.z) * w, __ATOMIC_RELAXED, __HIP_MEMORY_SCOPE_AGENT);
    __hip_atomic_fetch_add(p + 3, (av.w - bv.w) * w, __ATOMIC_RELAXED, __HIP_MEMORY_SCOPE_AGENT);
}

// sums[batch[n]] += h[n]; cnt[batch[n]] += 1
__global__ void pool_kernel(const float* __restrict__ h,
                            const int* __restrict__ batch,
                            float* __restrict__ sums, float* __restrict__ cnt) {
    int tid = blockIdx.x * blockDim.x + threadIdx.x;
    if (tid >= NNODES * 16) return;
    int n = tid >> 4;
    int j = (tid & 15) * 4;
    int g = batch[n];
    float4 v = *(const float4*)(h + n * HDIM + j);
    float* p = sums + g * HDIM + j;
    __hip_atomic_fetch_add(p + 0, v.x, __ATOMIC_RELAXED, __HIP_MEMORY_SCOPE_AGENT);
    __hip_atomic_fetch_add(p + 1, v.y, __ATOMIC_RELAXED, __HIP_MEMORY_SCOPE_AGENT);
    __hip_atomic_fetch_add(p + 2, v.z, __ATOMIC_RELAXED, __HIP_MEMORY_SCOPE_AGENT);
    __hip_atomic_fetch_add(p + 3, v.w, __ATOMIC_RELAXED, __HIP_MEMORY_SCOPE_AGENT);
    if ((tid & 15) == 0)
        __hip_atomic_fetch_add(cnt + g, 1.0f, __ATOMIC_RELAXED, __HIP_MEMORY_SCOPE_AGENT);
}

// graph_x = sums/cnt; out = relu(graph_x@Wf1+bf1)@Wf2+bf2.  One block per graph.
__global__ void __launch_bounds__(128)
head_kernel(const float* __restrict__ sums, const float* __restrict__ cnt,
            const float* __restrict__ Wf1, const float* __restrict__ bf1,
            const float* __restrict__ Wf2, const float* __restrict__ bf2,
            float* __restrict__ out) {
    __shared__ float gx[HDIM];
    __shared__ float hid[2 * HDIM];
    int g = blockIdx.x, t = threadIdx.x;
    if (t < HDIM) gx[t] = sums[g * HDIM + t] / fmaxf(cnt[g], 1.0f);
    __syncthreads();
    float acc = bf1[t];
#pragma unroll
    for (int k = 0; k < HDIM; ++k) acc += gx[k] * Wf1[k * 2 * HDIM + t];
    hid[t] = fmaxf(acc, 0.0f);
    __syncthreads();
    if (t < NCLASS) {
        float o = bf2[t];
        for (int k = 0; k < 2 * HDIM; ++k) o += hid[k] * Wf2[k * NCLASS + t];
        out[g * NCLASS + t] = o;
    }
}

extern "C" void kernel_launch(void* const* d_in, const int* in_sizes, int n_in,
                              void* d_out, int out_size, void* d_ws, size_t ws_size,
                              hipStream_t stream) {
    const float* x      = (const float*)d_in[0];
    const int*   ei     = (const int*)d_in[1];     // [2, E]
    const int*   batch  = (const int*)d_in[2];     // [N]
    const float* ea     = (const float*)d_in[3];   // [E]
    const float* W_emb  = (const float*)d_in[4];
    const float* b_emb  = (const float*)d_in[5];
    const float* W1     = (const float*)d_in[6];   // [L,64,64]
    const float* b1     = (const float*)d_in[7];   // [L,64]
    const float* W2     = (const float*)d_in[8];
    const float* W3     = (const float*)d_in[9];
    const float* b3     = (const float*)d_in[10];
    const float* Wf1    = (const float*)d_in[11];
    const float* bf1    = (const float*)d_in[12];
    const float* Wf2    = (const float*)d_in[13];
    const float* bf2    = (const float*)d_in[14];
    float* out = (float*)d_out;

    const int NH = NNODES * HDIM;
    const int WN = HDIM * HDIM;          // 4096
    float* a    = (float*)d_ws;          // [N,H]
    float* bbuf = a + NH;                // [N,H]
    float* agg  = bbuf + NH;             // [N,H]
    float* h    = agg + NH;              // [N,H]
    float* sums = h + NH;                // [G,H]
    float* cnt  = sums + NGRAPHS * HDIM; // [G]
    float* wt   = cnt + NGRAPHS;         // 9 transposed [64,64] weights

    const int B = 256;
    const int tiles  = (NNODES + 15) / 16;            // 16-row WMMA tiles
    const int gblk   = (tiles * 32 + B - 1) / B;      // 8 waves per block
    const int eblk   = (NEDGES * 16 + B - 1) / B;
    const int nhblk  = (NH + B - 1) / B;
    const int poolb  = (NNODES * 16 + B - 1) / B;

    transpose_w_kernel<<<(LAYERS * 3 * WN + B - 1) / B, B, 0, stream>>>(W1, W2, W3, wt);
    embed_kernel<<<nhblk, B, 0, stream>>>(x, W_emb, b_emb, h);

    for (int l = 0; l < LAYERS; ++l) {
        const float* wt1 = wt + (l * 3 + 0) * WN;
        const float* wt2 = wt + (l * 3 + 1) * WN;
        const float* wt3 = wt + (l * 3 + 2) * WN;
        const float* b1l = b1 + l * HDIM;
        const float* b3l = b3 + l * HDIM;

        zero_kernel<<<nhblk, B, 0, stream>>>(agg, NH);
        gemm2_kernel<<<gblk, B, 0, stream>>>(h, wt1, b1l, wt2, a, bbuf);
        edge_kernel<<<eblk, B, 0, stream>>>(ei, ea, a, bbuf, agg);
        gemm3_relu_kernel<<<gblk, B, 0, stream>>>(h, wt3, b3l, agg);
    }

    zero_kernel<<<(NGRAPHS * HDIM + NGRAPHS + B - 1) / B, B, 0, stream>>>(sums, NGRAPHS * HDIM + NGRAPHS);
    pool_kernel<<<poolb, B, 0, stream>>>(h, batch, sums, cnt);
    head_kernel<<<NGRAPHS, 128, 0, stream>>>(sums, cnt, Wf1, bf1, Wf2, bf2, out);
}